// SpatialFeatureExtractor_59734405153026
// MI455X (gfx1250) — compile-verified
//
#include <hip/hip_runtime.h>
#include <hip/hip_bf16.h>
#include <math.h>

// CDNA5 / gfx1250: fp32 matrix pipe, one workgroup per (b,f) covariance matrix.
typedef __attribute__((ext_vector_type(2))) float v2f;
typedef __attribute__((ext_vector_type(8))) float v8f;

#define C_DIM   64
#define T_DIM   1000
#define CHUNK   100           // 10 chunks of 64x100 fp32 streamed through LDS
#define NCHUNK  (T_DIM / CHUNK)
#define XPAD    101           // odd stride -> conflict-free LDS banks
#define APAD    65
#define NSWEEPS 14            // fixed sweep count (deterministic)
#define EIG_EPS 1e-10f

__device__ __forceinline__ void jacobi_pair(int rd, int i, int& p, int& q) {
    // round-robin tournament: 64 players, 63 rounds, 32 disjoint pairs/round
    if (i == 0) { p = 63; q = rd; }
    else        { p = (rd + i) % 63; q = (rd - i + 63) % 63; }
}

__global__ __launch_bounds__(256)
void spd_logm_kernel(const float* __restrict__ x, float* __restrict__ out) {
    __shared__ float sA[C_DIM * APAD];   // covariance -> Jacobi working matrix
    __shared__ float sV[C_DIM * APAD];   // eigenvectors
    __shared__ float sX[C_DIM * XPAD];   // streamed data chunk; reused as W = V*diag(log e)
    __shared__ float sR[C_DIM];          // row sums; reused as log-eigenvalues
    __shared__ float sC[32], sS[32];     // per-round rotation cos/sin

    const int tid  = threadIdx.x;
    const int lane = tid & 31;
    const int lhi  = lane >> 4;          // K-half selector (A/B frags)
    const int llo  = lane & 15;
    const int wave = tid >> 5;
    const size_t bf = blockIdx.x;
    const float* xb = x + bf * (size_t)(C_DIM * T_DIM);

    // 4x4 grid of 16x16 output tiles; wave w owns tiles 2w and 2w+1
    const int tt0 = wave * 2;
    const int tm0 = ((tt0    ) >> 2) * 16, tn0 = ((tt0    ) & 3) * 16;
    const int tm1 = ((tt0 + 1) >> 2) * 16, tn1 = ((tt0 + 1) & 3) * 16;

    v8f acc0 = {}; v8f acc1 = {};
    float rsum = 0.0f;                   // row-sum partial (threads 0..63)

    // ---------- Phase 1: Gram matrix G = X*X^T via V_WMMA_F32_16X16X4_F32 ----------
    for (int ch = 0; ch < NCHUNK; ++ch) {
        const int t0 = ch * CHUNK;
        for (int i = 0; i < (C_DIM * CHUNK) / 256; ++i) {
            int L = tid + i * 256;
            int c = L / CHUNK, k = L % CHUNK;
            sX[c * XPAD + k] = xb[c * T_DIM + t0 + k];
        }
        __syncthreads();

        if (ch + 1 < NCHUNK) {           // global_prefetch_b8 of next chunk
            int L = tid * ((C_DIM * CHUNK) / 256);
            int c = L / CHUNK, k = L % CHUNK;
            __builtin_prefetch(&xb[c * T_DIM + t0 + CHUNK + k], 0, 0);
        }

        if (tid < C_DIM) {               // deterministic row-sum accumulation
            float s = 0.0f;
            for (int k = 0; k < CHUNK; ++k) s += sX[tid * XPAD + k];
            rsum += s;
        }

        for (int k = 0; k < CHUNK; k += 4) {
            const int kb = k + lhi * 2;
            v2f a0, b0, a1, b1;
            a0.x = sX[(tm0 + llo) * XPAD + kb];     a0.y = sX[(tm0 + llo) * XPAD + kb + 1];
            b0.x = sX[(tn0 + llo) * XPAD + kb];     b0.y = sX[(tn0 + llo) * XPAD + kb + 1];
            acc0 = __builtin_amdgcn_wmma_f32_16x16x4_f32(false, a0, false, b0, (short)0, acc0, false, false);
            a1.x = sX[(tm1 + llo) * XPAD + kb];     a1.y = sX[(tm1 + llo) * XPAD + kb + 1];
            b1.x = sX[(tn1 + llo) * XPAD + kb];     b1.y = sX[(tn1 + llo) * XPAD + kb + 1];
            acc1 = __builtin_amdgcn_wmma_f32_16x16x4_f32(false, a1, false, b1, (short)0, acc1, false, false);
        }
        __syncthreads();
    }

    if (tid < C_DIM) sR[tid] = rsum;
    __syncthreads();

    // cov = (G - s s^T / T) / (T-1), scatter C-fragments into sA
    {
        const float invT = 1.0f / (float)T_DIM;
        const float invTm1 = 1.0f / (float)(T_DIM - 1);
        const float* a0p = (const float*)&acc0;
        const float* a1p = (const float*)&acc1;
        for (int r = 0; r < 8; ++r) {
            int M0 = tm0 + r + lhi * 8, N0 = tn0 + llo;
            sA[M0 * APAD + N0] = (a0p[r] - sR[M0] * sR[N0] * invT) * invTm1;
            int M1 = tm1 + r + lhi * 8, N1 = tn1 + llo;
            sA[M1 * APAD + N1] = (a1p[r] - sR[M1] * sR[N1] * invT) * invTm1;
        }
    }
    for (int i = tid; i < C_DIM * C_DIM; i += 256) {
        int m = i >> 6, n = i & 63;
        sV[m * APAD + n] = (m == n) ? 1.0f : 0.0f;
    }
    __syncthreads();

    // ---------- Phase 2: cyclic Jacobi eigensolver (fixed sweeps) ----------
    for (int sweep = 0; sweep < NSWEEPS; ++sweep) {
        for (int rd = 0; rd < 63; ++rd) {
            if (tid < 32) {
                int p, q; jacobi_pair(rd, tid, p, q);
                float app = sA[p * APAD + p];
                float aqq = sA[q * APAD + q];
                float apq = sA[p * APAD + q];
                float c = 1.0f, s = 0.0f;
                if (fabsf(apq) > 1e-30f) {
                    float tau = (aqq - app) / (2.0f * apq);
                    float t = copysignf(1.0f / (fabsf(tau) + sqrtf(tau * tau + 1.0f)), tau);
                    c = rsqrtf(t * t + 1.0f);
                    s = t * c;
                }
                sC[tid] = c; sS[tid] = s;
            }
            __syncthreads();
            // A <- J^T A  (disjoint row pairs; parallel over (pair, column))
            for (int w = tid; w < 32 * 64; w += 256) {
                int i = w >> 6, j = w & 63;
                int p, q; jacobi_pair(rd, i, p, q);
                float c = sC[i], s = sS[i];
                float ap = sA[p * APAD + j], aq = sA[q * APAD + j];
                sA[p * APAD + j] = c * ap - s * aq;
                sA[q * APAD + j] = s * ap + c * aq;
            }
            __syncthreads();
            // A <- A J ; V <- V J  (disjoint column pairs; parallel over (pair, row))
            for (int w = tid; w < 32 * 64; w += 256) {
                int i = w >> 6, j = w & 63;
                int p, q; jacobi_pair(rd, i, p, q);
                float c = sC[i], s = sS[i];
                float ap = sA[j * APAD + p], aq = sA[j * APAD + q];
                sA[j * APAD + p] = c * ap - s * aq;
                sA[j * APAD + q] = s * ap + c * aq;
                float vp = sV[j * APAD + p], vq = sV[j * APAD + q];
                sV[j * APAD + p] = c * vp - s * vq;
                sV[j * APAD + q] = s * vp + c * vq;
            }
            __syncthreads();
        }
    }

    // ---------- Phase 3: logm = (V diag(log e)) V^T via WMMA ----------
    if (tid < C_DIM) sR[tid] = logf(fmaxf(sA[tid * APAD + tid], EIG_EPS));
    __syncthreads();
    for (int i = tid; i < C_DIM * C_DIM; i += 256) {   // W into sX (reuse, stride APAD)
        int m = i >> 6, k = i & 63;
        sX[m * APAD + k] = sV[m * APAD + k] * sR[k];
    }
    __syncthreads();

    v8f r0 = {}; v8f r1 = {};
    for (int k = 0; k < C_DIM; k += 4) {
        const int kb = k + lhi * 2;
        v2f a0, b0, a1, b1;
        a0.x = sX[(tm0 + llo) * APAD + kb];     a0.y = sX[(tm0 + llo) * APAD + kb + 1];
        b0.x = sV[(tn0 + llo) * APAD + kb];     b0.y = sV[(tn0 + llo) * APAD + kb + 1];   // B = V^T
        r0 = __builtin_amdgcn_wmma_f32_16x16x4_f32(false, a0, false, b0, (short)0, r0, false, false);
        a1.x = sX[(tm1 + llo) * APAD + kb];     a1.y = sX[(tm1 + llo) * APAD + kb + 1];
        b1.x = sV[(tn1 + llo) * APAD + kb];     b1.y = sV[(tn1 + llo) * APAD + kb + 1];
        r1 = __builtin_amdgcn_wmma_f32_16x16x4_f32(false, a1, false, b1, (short)0, r1, false, false);
    }

    // upper-triangular vectorization: idx(M,N) = 64M - M(M-1)/2 + (N-M)
    float* ob = out + bf * (size_t)2080;
    const float* r0p = (const float*)&r0;
    const float* r1p = (const float*)&r1;
    for (int r = 0; r < 8; ++r) {
        int M0 = tm0 + r + lhi * 8, N0 = tn0 + llo;
        if (N0 >= M0) ob[M0 * C_DIM - (M0 * (M0 - 1)) / 2 + (N0 - M0)] = r0p[r];
        int M1 = tm1 + r + lhi * 8, N1 = tn1 + llo;
        if (N1 >= M1) ob[M1 * C_DIM - (M1 * (M1 - 1)) / 2 + (N1 - M1)] = r1p[r];
    }
}

extern "C" void kernel_launch(void* const* d_in, const int* in_sizes, int n_in,
                              void* d_out, int out_size, void* d_ws, size_t ws_size,
                              hipStream_t stream) {
    (void)n_in; (void)out_size; (void)d_ws; (void)ws_size;
    const float* x = (const float*)d_in[0];
    float* out = (float*)d_out;
    const int nbf = in_sizes[0] / (C_DIM * T_DIM);     // 256*8 = 2048 matrices
    hipLaunchKernelGGL(spd_logm_kernel, dim3(nbf), dim3(256), 0, stream, x, out);
}